// Synthesize_view_scared_1889785610446
// MI455X (gfx1250) — compile-verified
//
#include <hip/hip_runtime.h>
#include <hip/hip_bf16.h>

// ---------------------------------------------------------------------------
// Synthesize view (SCARED): disp->depth, backproject, translate, project,
// bilinear grid_sample (border padding). B=8, H=1024, W=1280.
//
// Strategy (MI455X, gfx1250):
//  * Hot loop is HBM-bound (~0.3 GB total -> ~13 us at 23.3 TB/s). One thread
//    per output pixel; 4 gather addresses + weights shared across 3 channels.
//  * All matrix algebra (K_tgt * inv(K_src), K_tgt * t) hoisted into a tiny
//    per-batch setup kernel that uses V_WMMA_F32_16X16X4_F32 (the one genuine
//    matrix op in this workload) and writes 12 floats/batch to workspace.
//  * Per-batch constants loaded with a blockIdx.y-uniform address -> scalar
//    loads. Streaming disp read + output writes use non-temporal hints so the
//    img gathers keep the caches.
// ---------------------------------------------------------------------------

#define BATCH 8
#define IMG_H 1024
#define IMG_W 1280
#define HW (IMG_H * IMG_W)

typedef __attribute__((ext_vector_type(2))) float v2f;
typedef __attribute__((ext_vector_type(8))) float v8f;

// ---------------------------------------------------------------------------
// Setup kernel: one wave (32 lanes) per batch.
// Computes Mv[3][4] = [ K_tgt3x3 * inv(K_src3x3) | K_tgt3x3 * t ]
// via WMMA f32 16x16x4:  A(16x4) = K_tgt padded, B(4x16) = [invK | t] padded.
// Writes 12 floats per batch into ws.
// ---------------------------------------------------------------------------
__global__ __launch_bounds__(32)
void vs_setup_kernel(const float* __restrict__ source_K,   // [B,4,4]
                     const float* __restrict__ target_K,   // [B,4,4]
                     const float* __restrict__ trans_vec,  // [B,3]
                     float* __restrict__ ws)               // [B,12]
{
    const int b    = blockIdx.x;
    const int lane = threadIdx.x;

    const float* S  = source_K + b * 16;
    const float* Tk = target_K + b * 16;
    const float* tv = trans_vec + b * 3;

    // --- inverse of source_K's 3x3 block (uniform across lanes; SALU-ish) ---
    float s00 = S[0], s01 = S[1], s02 = S[2];
    float s10 = S[4], s11 = S[5], s12 = S[6];
    float s20 = S[8], s21 = S[9], s22 = S[10];
    float c00 =  (s11 * s22 - s12 * s21);
    float c01 = -(s10 * s22 - s12 * s20);
    float c02 =  (s10 * s21 - s11 * s20);
    float det = s00 * c00 + s01 * c01 + s02 * c02;
    float rd  = 1.0f / det;
    // inv[i][j] = cof[j][i] / det
    float i00 =  (s11 * s22 - s12 * s21) * rd;
    float i01 = -(s01 * s22 - s02 * s21) * rd;
    float i02 =  (s01 * s12 - s02 * s11) * rd;
    float i10 = -(s10 * s22 - s12 * s20) * rd;
    float i11 =  (s00 * s22 - s02 * s20) * rd;
    float i12 = -(s00 * s12 - s02 * s10) * rd;
    float i20 =  (s10 * s21 - s11 * s20) * rd;
    float i21 = -(s00 * s21 - s01 * s20) * rd;
    float i22 =  (s00 * s11 - s01 * s10) * rd;
    float t0 = tv[0], t1 = tv[1], t2 = tv[2];

#if __has_builtin(__builtin_amdgcn_wmma_f32_16x16x4_f32)
    // ---- A matrix 16x4 f32 layout (2 VGPRs):
    //   a.x : lanes 0-15 -> A[M=lane][K=0]   ; lanes 16-31 -> A[M=lane-16][K=2]
    //   a.y : lanes 0-15 -> A[M=lane][K=1]   ; lanes 16-31 -> A[M=lane-16][K=3]
    const int  m    = lane & 15;
    const bool lo   = lane < 16;
    const bool mok  = m < 3;
    const int  msafe = mok ? m : 0;
    float tk_c0 = Tk[msafe * 4 + 0];
    float tk_c1 = Tk[msafe * 4 + 1];
    float tk_c2 = Tk[msafe * 4 + 2];
    v2f a;
    a.x = mok ? (lo ? tk_c0 : tk_c2) : 0.0f;   // K=0 / K=2
    a.y = mok ? (lo ? tk_c1 : 0.0f)  : 0.0f;   // K=1 / K=3 (col 3 = 0)

    // ---- B matrix 4x16 f32 layout (2 VGPRs):
    //   b.x : lanes 0-15 -> B[K=0][N=lane] ; lanes 16-31 -> B[K=2][N=lane-16]
    //   b.y : lanes 0-15 -> B[K=1][N=lane] ; lanes 16-31 -> B[K=3][N=lane-16]
    // B = [ invK (3x3) | t (col 3) ], row 3 = 0.
    const int n = lane & 15;
    float r0 = (n == 0) ? i00 : (n == 1) ? i01 : (n == 2) ? i02 : (n == 3) ? t0 : 0.0f;
    float r1 = (n == 0) ? i10 : (n == 1) ? i11 : (n == 2) ? i12 : (n == 3) ? t1 : 0.0f;
    float r2 = (n == 0) ? i20 : (n == 1) ? i21 : (n == 2) ? i22 : (n == 3) ? t2 : 0.0f;
    v2f bb;
    bb.x = lo ? r0 : r2;     // K=0 / K=2
    bb.y = lo ? r1 : 0.0f;   // K=1 / K=3 (row 3 = 0)

    v8f acc = {};
    // args: (neg_a, A, neg_b, B, c_mod, C, reuse_a, reuse_b)
    acc = __builtin_amdgcn_wmma_f32_16x16x4_f32(false, a, false, bb,
                                                (short)0, acc, false, false);

    // D 16x16 f32: d[r] lanes 0-15 hold D[M=r][N=lane]. Need rows 0..2, cols 0..3.
    if (lane < 4) {
        ws[b * 12 + 0 * 4 + lane] = acc[0];
        ws[b * 12 + 1 * 4 + lane] = acc[1];
        ws[b * 12 + 2 * 4 + lane] = acc[2];
    }
#else
    // Scalar fallback: lane 0 computes Mv = [K_tgt*invK | K_tgt*t] directly.
    if (lane == 0) {
        float K[3][3] = {{Tk[0], Tk[1], Tk[2]},
                         {Tk[4], Tk[5], Tk[6]},
                         {Tk[8], Tk[9], Tk[10]}};
        float IV[3][3] = {{i00, i01, i02}, {i10, i11, i12}, {i20, i21, i22}};
        float tt[3] = {t0, t1, t2};
        for (int r = 0; r < 3; ++r) {
            for (int c = 0; c < 3; ++c) {
                float acc = 0.0f;
                for (int k = 0; k < 3; ++k) acc += K[r][k] * IV[k][c];
                ws[b * 12 + r * 4 + c] = acc;
            }
            float vv = 0.0f;
            for (int k = 0; k < 3; ++k) vv += K[r][k] * tt[k];
            ws[b * 12 + r * 4 + 3] = vv;
        }
    }
#endif
}

// ---------------------------------------------------------------------------
// Main kernel: one thread per output pixel, all 3 channels.
// grid = (HW/256, B); blockIdx.y = batch -> constants load is wave-uniform.
// ---------------------------------------------------------------------------
__global__ __launch_bounds__(256)
void vs_warp_kernel(const float* __restrict__ img,        // [B,3,H,W]
                    const float* __restrict__ pred_disp,  // [B,1,H,W]
                    const float* __restrict__ consts,     // [B,12] from ws
                    float* __restrict__ out)              // [B,3,H,W]
{
    const int tid = blockIdx.x * blockDim.x + threadIdx.x;
    if (tid >= HW) return;
    const int b = blockIdx.y;

    // Per-batch projection constants (uniform address -> scalar loads).
    const float* C = consts + b * 12;
    const float m00 = C[0],  m01 = C[1],  m02 = C[2],  v0 = C[3];
    const float m10 = C[4],  m11 = C[5],  m12 = C[6],  v1 = C[7];
    const float m20 = C[8],  m21 = C[9],  m22 = C[10], v2 = C[11];

    const int py = tid / IMG_W;
    const int px = tid - py * IMG_W;
    const float fx = (float)px;
    const float fy = (float)py;

    // disp -> depth  (min_disp = 1/255, max_disp = 1/10)
    const float disp = __builtin_nontemporal_load(pred_disp + (size_t)b * HW + tid);
    const float min_disp = 1.0f / 255.0f;
    const float max_disp = 1.0f / 10.0f;
    const float scaled   = min_disp + (max_disp - min_disp) * disp;
    const float depth    = 1.0f / scaled;

    // cp = depth * (M @ [x,y,1]) + v ;  pc = cp.xy / (cp.z + EPS)
    const float nx  = depth * (m00 * fx + m01 * fy + m02) + v0;
    const float ny  = depth * (m10 * fx + m11 * fy + m12) + v1;
    const float den = depth * (m20 * fx + m21 * fy + m22) + v2 + 1e-7f;
    const float rden = 1.0f / den;
    const float pcx = nx * rden;
    const float pcy = ny * rden;

    // grid = (pc/[W-1,H-1] - 0.5)*2 ; sample x = (gx+1)*W/2 - 0.5
    const float sx = pcx * ((float)IMG_W / (float)(IMG_W - 1)) - 0.5f;
    const float sy = pcy * ((float)IMG_H / (float)(IMG_H - 1)) - 0.5f;

    // bilinear, border clamp (torch grid_sample defaults)
    const float x0f = floorf(sx);
    const float y0f = floorf(sy);
    const float wx = sx - x0f;
    const float wy = sy - y0f;
    // clamp in float first so int conversion is well-defined, then int clamp
    const float x0c = fminf(fmaxf(x0f, -2.0f), (float)IMG_W + 2.0f);
    const float y0c = fminf(fmaxf(y0f, -2.0f), (float)IMG_H + 2.0f);
    int x0 = (int)x0c, y0 = (int)y0c;
    int x1 = x0 + 1,  y1 = y0 + 1;
    x0 = min(max(x0, 0), IMG_W - 1);
    x1 = min(max(x1, 0), IMG_W - 1);
    y0 = min(max(y0, 0), IMG_H - 1);
    y1 = min(max(y1, 0), IMG_H - 1);

    const float w11 = wx * wy;
    const float w01 = wx - w11;          // wx*(1-wy)
    const float w10 = wy - w11;          // (1-wx)*wy
    const float w00 = 1.0f - wx - w10;   // (1-wx)*(1-wy)

    const int o00 = y0 * IMG_W + x0;
    const int o01 = y0 * IMG_W + x1;
    const int o10 = y1 * IMG_W + x0;
    const int o11 = y1 * IMG_W + x1;

    const size_t ibase = (size_t)b * 3 * HW;
#pragma unroll
    for (int ch = 0; ch < 3; ++ch) {
        const float* p = img + ibase + (size_t)ch * HW;
        const float val = p[o00] * w00 + p[o01] * w01
                        + p[o10] * w10 + p[o11] * w11;
        __builtin_nontemporal_store(val, out + ibase + (size_t)ch * HW + tid);
    }
}

// ---------------------------------------------------------------------------
extern "C" void kernel_launch(void* const* d_in, const int* in_sizes, int n_in,
                              void* d_out, int out_size, void* d_ws, size_t ws_size,
                              hipStream_t stream) {
    const float* img       = (const float*)d_in[0];  // [B,3,H,W]
    const float* pred_disp = (const float*)d_in[1];  // [B,1,H,W]
    const float* source_K  = (const float*)d_in[2];  // [B,4,4]
    const float* target_K  = (const float*)d_in[3];  // [B,4,4]
    const float* trans_vec = (const float*)d_in[4];  // [B,3]
    float* out = (float*)d_out;
    float* ws  = (float*)d_ws;                       // [B,12] constants

    const int batch = in_sizes[1] / HW;              // = 8

    vs_setup_kernel<<<dim3(batch), dim3(32), 0, stream>>>(
        source_K, target_K, trans_vec, ws);

    dim3 grid((HW + 255) / 256, batch);
    vs_warp_kernel<<<grid, dim3(256), 0, stream>>>(img, pred_disp, ws, out);
}